// Attention_62835371540497
// MI455X (gfx1250) — compile-verified
//
#include <hip/hip_runtime.h>
#include <hip/hip_bf16.h>

// Problem constants (from reference)
#define B_    8
#define N_    1024
#define C_    768
#define H_    12
#define HD_   64
#define NB_   49      // real bucket count
#define NBP_  64      // padded bucket columns in lk/lq/ab
#define ROWS_ 32      // attention row-stripe per workgroup
#define BSTR_ 66      // LDS stride for transposed weight panel (even, !=64)
#define VCH_  128     // V staging chunk (rows of V per LDS stage)
#define VSTR_ 130     // LDS stride for transposed V chunk (even, !=128)

typedef float v2f __attribute__((ext_vector_type(2)));
typedef float v8f __attribute__((ext_vector_type(8)));

static __device__ __forceinline__ v8f wmma4(v2f a, v2f b, v8f c) {
  // V_WMMA_F32_16X16X4_F32 : D(16x16 f32) = A(16x4 f32) * B(4x16 f32) + C
  return __builtin_amdgcn_wmma_f32_16x16x4_f32(false, a, false, b, (short)0, c,
                                               false, false);
}
static __device__ __forceinline__ v2f ld2(const float* p) {
  return *(const v2f*)p;
}

// ---------------------------------------------------------------------------
// Kernel 1: QKV GEMM.  x[8192,768] @ W_qkv[768,2304] scattered into
// q/k/v laid out [B,H,N,HD].
// Block = 256 thr (8 waves) -> 128 rows x 64 cols.  Weight panel staged
// through LDS transposed; per-wmma B operand is one aligned ds_load_b64.
// grid = (64 row-blocks, 36 col-groups)
// ---------------------------------------------------------------------------
__global__ __launch_bounds__(256) void k_qkv(const float* __restrict__ x,
                                             const float* __restrict__ wqkv,
                                             float* __restrict__ qb,
                                             float* __restrict__ kb,
                                             float* __restrict__ vb) {
  __shared__ float Bt[64 * BSTR_];          // Bt[n][k], k-chunk of 64

  const int wave  = threadIdx.x >> 5;
  const int lane  = threadIdx.x & 31;
  const int l15   = lane & 15;
  const int lhalf = lane >> 4;

  const int ngrp = blockIdx.y;              // 64-col group of W_qkv (0..35)
  const int nb   = ngrp * 64;
  const int row  = blockIdx.x * 128 + wave * 16 + l15;
  const float* xr = x + (size_t)row * C_;

  v8f acc[4];
  acc[0] = (v8f)0.f; acc[1] = (v8f)0.f; acc[2] = (v8f)0.f; acc[3] = (v8f)0.f;

  for (int k0 = 0; k0 < C_; k0 += 64) {
    // ---- stage W panel (64k x 64n) transposed into LDS ----
    __syncthreads();
#pragma unroll
    for (int r = 0; r < 4; ++r) {
      const int idx = threadIdx.x + 256 * r;     // 0..1023
      const int k   = idx >> 4;                  // 0..63
      const int nq  = (idx & 15) * 4;            // 0,4,..,60
      const float4 w = *(const float4*)(wqkv + (size_t)(k0 + k) * (3 * C_) + nb + nq);
      Bt[(nq + 0) * BSTR_ + k] = w.x;
      Bt[(nq + 1) * BSTR_ + k] = w.y;
      Bt[(nq + 2) * BSTR_ + k] = w.z;
      Bt[(nq + 3) * BSTR_ + k] = w.w;
      if (k0 + 64 < C_)
        __builtin_prefetch(wqkv + (size_t)(k0 + 64 + k) * (3 * C_) + nb + nq, 0, 1);
    }
    __syncthreads();

    // ---- compute on the staged panel ----
#pragma unroll
    for (int kk = 0; kk < 64; kk += 4) {
      const int kbi = kk + 2 * lhalf;
      v2f a = ld2(xr + k0 + kbi);
#pragma unroll
      for (int t = 0; t < 4; ++t) {
        v2f bf = ld2(&Bt[(t * 16 + l15) * BSTR_ + kbi]);
        acc[t] = wmma4(a, bf, acc[t]);
      }
    }
  }

  // ngrp in [0,36): which = ngrp/12 selects q/k/v, h = ngrp%12, d = t*16+l15
  const int which = ngrp / H_;
  const int h     = ngrp % H_;
  float* dst = (which == 0) ? qb : (which == 1) ? kb : vb;
#pragma unroll
  for (int t = 0; t < 4; ++t) {
    const int d = t * 16 + l15;
#pragma unroll
    for (int vv = 0; vv < 8; ++vv) {
      const int r    = blockIdx.x * 128 + wave * 16 + vv + 8 * lhalf;
      const int btok = r >> 10;
      const int tok  = r & (N_ - 1);
      dst[(((size_t)btok * H_ + h) * N_ + tok) * HD_ + d] = acc[t][vv];
    }
  }
}

// ---------------------------------------------------------------------------
// Kernel 2: bucket-table GEMMs.
// lk[bh,i,buck] = q[bh,i,:] . table_k[h,buck,:]
// lq[bh,j,buck] = (k[bh,j,:]*scale) . table_q[h,buck,:]
// Padded to 64 bucket columns (cols >= 49 are zero).
// ---------------------------------------------------------------------------
__global__ __launch_bounds__(256) void k_tables(const float* __restrict__ qb,
                                                const float* __restrict__ kb,
                                                const float* __restrict__ tq,
                                                const float* __restrict__ tk,
                                                float* __restrict__ lkb,
                                                float* __restrict__ lqb) {
  const int wave  = threadIdx.x >> 5;
  const int lane  = threadIdx.x & 31;
  const int l15   = lane & 15;
  const int lhalf = lane >> 4;

  const int unit = blockIdx.x * 8 + wave;   // 96 bh * 64 mtiles
  const int bh   = unit >> 6;
  const int mt   = unit & 63;
  const int h    = bh % H_;

  const size_t rowbase = ((size_t)bh * N_ + mt * 16 + l15) * HD_;
  const v2f z2 = (v2f)0.f;

  v8f ck[4], cq[4];
#pragma unroll
  for (int t = 0; t < 4; ++t) { ck[t] = (v8f)0.f; cq[t] = (v8f)0.f; }

#pragma unroll
  for (int kk = 0; kk < HD_; kk += 4) {
    const int kbi = kk + 2 * lhalf;
    v2f aq = ld2(qb + rowbase + kbi);
    v2f ak = ld2(kb + rowbase + kbi);
    ak = ak * 0.125f;                       // scale = HD^-0.5
#pragma unroll
    for (int t = 0; t < 4; ++t) {
      const int n = t * 16 + l15;           // bucket index
      v2f btk = (n < NB_) ? ld2(tk + ((size_t)h * NB_ + n) * HD_ + kbi) : z2;
      v2f btq = (n < NB_) ? ld2(tq + ((size_t)h * NB_ + n) * HD_ + kbi) : z2;
      ck[t] = wmma4(aq, btk, ck[t]);
      cq[t] = wmma4(ak, btq, cq[t]);
    }
  }

#pragma unroll
  for (int t = 0; t < 4; ++t) {
    const int col = t * 16 + l15;
#pragma unroll
    for (int vv = 0; vv < 8; ++vv) {
      const int r = mt * 16 + vv + 8 * lhalf;
      lkb[((size_t)bh * N_ + r) * NBP_ + col] = ck[t][vv];
      lqb[((size_t)bh * N_ + r) * NBP_ + col] = cq[t][vv];
    }
  }
}

// ---------------------------------------------------------------------------
// Kernel 3: fused attention for one (bh, 32-row stripe).
//  S = q@k^T + gather(lk, rb[i,j]) + gather(lq, rb[j,i])  (kept in LDS)
//  softmax rows; ab[i,buck] += P[i,j]; O = P@V + ab@table_v
// grid = (32 row-blocks, 96 bh), block = 256 (8 waves)
// dynamic LDS: S 32x1024 + ab 32x64 + Vt 64x130 = 172544 bytes (<320KB/WGP)
// ---------------------------------------------------------------------------
__global__ __launch_bounds__(256) void k_attn(const float* __restrict__ qb,
                                              const float* __restrict__ kb,
                                              const float* __restrict__ vb,
                                              const float* __restrict__ lkb,
                                              const float* __restrict__ lqb,
                                              const float* __restrict__ tv,
                                              const int*   __restrict__ rp,
                                              float* __restrict__ ob) {
  extern __shared__ float smem[];
  float* S  = smem;                              // [ROWS_][N_]
  float* ab = smem + ROWS_ * N_;                 // [ROWS_][NBP_]
  float* Vt = smem + ROWS_ * N_ + ROWS_ * NBP_;  // [HD_][VSTR_]

  const int wave  = threadIdx.x >> 5;
  const int lane  = threadIdx.x & 31;
  const int l15   = lane & 15;
  const int lhalf = lane >> 4;

  const int rblk = blockIdx.x;
  const int bh   = blockIdx.y;
  const int b    = bh / H_;
  const int h    = bh % H_;
  const int r0   = rblk * ROWS_;

  // ---- Step 1: scores + RPE gathers -> LDS -------------------------------
  // wave owns mtile = wave&1 (q fragments preloaded once), 16 K-tiles each
  {
    const int mt = wave & 1;
    const float* qrow = qb + ((size_t)bh * N_ + r0 + mt * 16 + l15) * HD_;
    v2f qf[16];
#pragma unroll
    for (int s = 0; s < 16; ++s) qf[s] = ld2(qrow + 4 * s + 2 * lhalf);

    for (int nt = (wave >> 1); nt < 64; nt += 4) {
      const float* krow = kb + ((size_t)bh * N_ + nt * 16 + l15) * HD_;
      v8f acc = (v8f)0.f;
#pragma unroll
      for (int s = 0; s < 16; ++s)
        acc = wmma4(qf[s], ld2(krow + 4 * s + 2 * lhalf), acc);

      const int n = nt * 16 + l15;               // column j
#pragma unroll
      for (int vv = 0; vv < 8; ++vv) {
        const int m   = mt * 16 + vv + 8 * lhalf;
        const int i   = r0 + m;
        const int bij = rp[i * N_ + n];
        const int bji = rp[n * N_ + i];
        float val = acc[vv]
                  + lkb[((size_t)bh * N_ + i) * NBP_ + bij]
                  + lqb[((size_t)bh * N_ + n) * NBP_ + bji];
        S[m * N_ + n] = val;
      }
    }
  }
  __syncthreads();

  // ---- Step 2: row softmax in LDS (4 rows per wave) ----------------------
  for (int rr = 0; rr < 4; ++rr) {
    const int m = wave * 4 + rr;
    float* row = S + m * N_;
    float mx = -3.4e38f;
    for (int j = lane; j < N_; j += 32) mx = fmaxf(mx, row[j]);
#pragma unroll
    for (int off = 16; off > 0; off >>= 1)
      mx = fmaxf(mx, __shfl_xor(mx, off, 32));
    float sum = 0.f;
    for (int j = lane; j < N_; j += 32) {
      float e = __expf(row[j] - mx);
      row[j] = e;
      sum += e;
    }
#pragma unroll
    for (int off = 16; off > 0; off >>= 1)
      sum += __shfl_xor(sum, off, 32);
    const float inv = 1.0f / sum;
    for (int j = lane; j < N_; j += 32) row[j] *= inv;
  }
  __syncthreads();

  // ---- Step 3: bucket scatter-add ab[i,buck] += P[i,j] -------------------
  for (int idx = threadIdx.x; idx < ROWS_ * NBP_; idx += 256) ab[idx] = 0.f;
  __syncthreads();
  for (int rr = 0; rr < 4; ++rr) {
    const int m = wave * 4 + rr;
    const int i = r0 + m;
    for (int j = lane; j < N_; j += 32) {
      const int bkt = rp[i * N_ + j];
      atomicAdd(&ab[m * NBP_ + bkt], S[m * N_ + j]);
    }
  }

  // ---- Step 4: O = P@V + ab@table_v, one 16x16 tile per wave -------------
  const int mt   = wave & 1;
  const int dt   = wave >> 1;                    // 4 groups of 16 over HD
  const int dcol = dt * 16 + l15;
  v8f acc = (v8f)0.f;

  for (int ch = 0; ch < N_; ch += VCH_) {        // P @ V, V staged in LDS
    __syncthreads();                             // Vt free / prev chunk done
    // stage V[ch..ch+127][0..63] transposed into Vt[d][j]
#pragma unroll
    for (int r = 0; r < 8; ++r) {
      const int idx = threadIdx.x + 256 * r;     // 0..2047
      const int j   = idx >> 4;                  // 0..127
      const int dq  = (idx & 15) * 4;            // 0,4,..,60
      const float4 w = *(const float4*)(vb + ((size_t)bh * N_ + ch + j) * HD_ + dq);
      Vt[(dq + 0) * VSTR_ + j] = w.x;
      Vt[(dq + 1) * VSTR_ + j] = w.y;
      Vt[(dq + 2) * VSTR_ + j] = w.z;
      Vt[(dq + 3) * VSTR_ + j] = w.w;
    }
    __syncthreads();
#pragma unroll
    for (int kk = 0; kk < VCH_; kk += 4) {
      const int kbi = kk + 2 * lhalf;
      v2f a  = ld2(&S[(mt * 16 + l15) * N_ + ch + kbi]);
      v2f bf = ld2(&Vt[dcol * VSTR_ + kbi]);
      acc = wmma4(a, bf, acc);
    }
  }

#pragma unroll
  for (int kk = 0; kk < NBP_; kk += 4) {         // ab @ table_v
    const int kbi = kk + 2 * lhalf;
    v2f a = ld2(&ab[(mt * 16 + l15) * NBP_ + kbi]);
    v2f bf;
    bf.x = (kbi     < NB_) ? tv[((size_t)h * NB_ + kbi)     * HD_ + dcol] : 0.f;
    bf.y = (kbi + 1 < NB_) ? tv[((size_t)h * NB_ + kbi + 1) * HD_ + dcol] : 0.f;
    acc = wmma4(a, bf, acc);
  }

#pragma unroll
  for (int vv = 0; vv < 8; ++vv) {
    const int m   = mt * 16 + vv + 8 * lhalf;
    const int tok = r0 + m;
    ob[(((size_t)b * N_ + tok) * H_ + h) * HD_ + dcol] = acc[vv];
  }
}

// ---------------------------------------------------------------------------
// Kernel 4: output projection. o[8192,768] @ W_proj[768,768] + b_proj
// Same LDS-staged structure as k_qkv.  grid = (64, 12)
// ---------------------------------------------------------------------------
__global__ __launch_bounds__(256) void k_proj(const float* __restrict__ ob,
                                              const float* __restrict__ wp,
                                              const float* __restrict__ bias,
                                              float* __restrict__ out) {
  __shared__ float Bt[64 * BSTR_];

  const int wave  = threadIdx.x >> 5;
  const int lane  = threadIdx.x & 31;
  const int l15   = lane & 15;
  const int lhalf = lane >> 4;

  const int ngrp = blockIdx.y;               // 0..11
  const int nb   = ngrp * 64;
  const int row  = blockIdx.x * 128 + wave * 16 + l15;
  const float* xr = ob + (size_t)row * C_;

  v8f acc[4];
  acc[0] = (v8f)0.f; acc[1] = (v8f)0.f; acc[2] = (v8f)0.f; acc[3] = (v8f)0.f;

  for (int k0 = 0; k0 < C_; k0 += 64) {
    __syncthreads();
#pragma unroll
    for (int r = 0; r < 4; ++r) {
      const int idx = threadIdx.x + 256 * r;
      const int k   = idx >> 4;
      const int nq  = (idx & 15) * 4;
      const float4 w = *(const float4*)(wp + (size_t)(k0 + k) * C_ + nb + nq);
      Bt[(nq + 0) * BSTR_ + k] = w.x;
      Bt[(nq + 1) * BSTR_ + k] = w.y;
      Bt[(nq + 2) * BSTR_ + k] = w.z;
      Bt[(nq + 3) * BSTR_ + k] = w.w;
      if (k0 + 64 < C_)
        __builtin_prefetch(wp + (size_t)(k0 + 64 + k) * C_ + nb + nq, 0, 1);
    }
    __syncthreads();

#pragma unroll
    for (int kk = 0; kk < 64; kk += 4) {
      const int kbi = kk + 2 * lhalf;
      v2f a = ld2(xr + k0 + kbi);
#pragma unroll
      for (int t = 0; t < 4; ++t) {
        v2f bf = ld2(&Bt[(t * 16 + l15) * BSTR_ + kbi]);
        acc[t] = wmma4(a, bf, acc[t]);
      }
    }
  }

#pragma unroll
  for (int t = 0; t < 4; ++t) {
    const int c0 = nb + t * 16 + l15;
    const float bv = bias[c0];
#pragma unroll
    for (int vv = 0; vv < 8; ++vv) {
      const int r = blockIdx.x * 128 + wave * 16 + vv + 8 * lhalf;
      out[(size_t)r * C_ + c0] = acc[t][vv] + bv;
    }
  }
}

// ---------------------------------------------------------------------------
extern "C" void kernel_launch(void* const* d_in, const int* in_sizes, int n_in,
                              void* d_out, int out_size, void* d_ws, size_t ws_size,
                              hipStream_t stream) {
  const float* x     = (const float*)d_in[0];
  const float* wqkv  = (const float*)d_in[1];
  const float* wproj = (const float*)d_in[2];
  const float* bproj = (const float*)d_in[3];
  const float* tq    = (const float*)d_in[4];
  const float* tk    = (const float*)d_in[5];
  const float* tv    = (const float*)d_in[6];
  const int*   rp    = (const int*)d_in[7];
  float*       out   = (float*)d_out;

  float* ws = (float*)d_ws;
  const size_t SZ = (size_t)B_ * H_ * N_ * HD_;   // 6,291,456 floats
  float* qb  = ws;
  float* kb  = ws + 1 * SZ;
  float* vb  = ws + 2 * SZ;
  float* lkb = ws + 3 * SZ;   // [B*H, N, 64]
  float* lqb = ws + 4 * SZ;   // [B*H, N, 64]
  float* ob  = ws + 5 * SZ;   // [B, N, H, HD] == [B*N, C]

  // K1: (64 row-blocks of 128, 36 col-groups of 64)
  k_qkv<<<dim3(64, 36), 256, 0, stream>>>(x, wqkv, qb, kb, vb);

  // K2: 96 bh * 64 mtiles = 6144 wave-units / 8 = 768 blocks
  k_tables<<<768, 256, 0, stream>>>(qb, kb, tq, tk, lkb, lqb);

  // K3: (32 row blocks, 96 bh), dynamic LDS = 172544 B (fits 320KB/WGP)
  const size_t lds = (size_t)(ROWS_ * N_ + ROWS_ * NBP_ + HD_ * VSTR_) * sizeof(float);
  k_attn<<<dim3(32, 96), 256, lds, stream>>>(qb, kb, vb, lkb, lqb, tv, rp, ob);

  // K4: (64 row-blocks of 128, 12 col-groups of 64)
  k_proj<<<dim3(64, 12), 256, 0, stream>>>(ob, wproj, bproj, out);

  (void)in_sizes; (void)n_in; (void)out_size; (void)ws_size;
}